// GNN_19464791785808
// MI455X (gfx1250) — compile-verified
//
#include <hip/hip_runtime.h>
#include <stdint.h>

// GCN 2-layer forward for MI455X (gfx1250, wave32).
// GEMM via V_WMMA_F32_16X16X4_F32 (f32-exact, GEMM is not the bottleneck);
// edge aggregation via coalesced float4 gathers + global_atomic_add_f32,
// which stays resident in the 192MB L2 (working set ~75MB).

typedef __attribute__((ext_vector_type(2))) float v2f;
typedef __attribute__((ext_vector_type(8))) float v8f;

// ---------------- degree / normalization ----------------

__global__ void deg_init_kernel(float* __restrict__ deg, int n) {
    int i = blockIdx.x * blockDim.x + threadIdx.x;
    if (i < n) deg[i] = 1.0f;                 // self-loop weight
}

__global__ void deg_acc_kernel(const long long* __restrict__ dst,
                               const float* __restrict__ w,
                               float* __restrict__ deg, int E) {
    int e = blockIdx.x * blockDim.x + threadIdx.x;
    if (e < E) atomicAdd(&deg[(int)dst[e]], w[e]);
}

__global__ void deg_rsqrt_kernel(float* __restrict__ deg, int n) {
    int i = blockIdx.x * blockDim.x + threadIdx.x;
    if (i < n) deg[i] = rsqrtf(deg[i]);       // deg >= 1 always (self loop)
}

// ---------------- WMMA GEMM: H[n, NT*16] = X[n, K] @ W[K, NT*16] ----------------
// One wave per 16-row stripe; NT column tiles of 16; K in steps of 4 via
// V_WMMA_F32_16X16X4_F32. A fragment reused across all NT column tiles.

template <int NT>
__global__ void gemm_wmma_kernel(const float* __restrict__ X,
                                 const float* __restrict__ W,
                                 float* __restrict__ H,
                                 int nRows, int K, int doRelu) {
    const int ncols = NT * 16;
    const int wave = threadIdx.x >> 5;        // wave32
    const int lane = threadIdx.x & 31;
    const int rowTile = blockIdx.x * (blockDim.x >> 5) + wave;
    const int rowBase = rowTile * 16;
    if (rowBase >= nRows) return;             // wave-uniform: EXEC stays all-1s

    const int m  = lane & 15;                 // row within tile (A) / col within tile (B)
    const int kp = (lane >> 4) << 1;          // K sub-pair: 0 or 2

    int row = rowBase + m;
    if (row >= nRows) row = nRows - 1;        // safe dup for ragged tail (stores guarded)

    v8f acc[NT] = {};

    for (int kb = 0; kb < K; kb += 4) {
        v2f a;
        float ax = X[(size_t)row * K + kb + kp];
        float ay = X[(size_t)row * K + kb + kp + 1];
        if (doRelu) { ax = fmaxf(ax, 0.0f); ay = fmaxf(ay, 0.0f); }
        a.x = ax; a.y = ay;
#pragma unroll
        for (int t = 0; t < NT; ++t) {
            const int col = t * 16 + m;
            v2f b;
            b.x = W[(kb + kp) * ncols + col];
            b.y = W[(kb + kp + 1) * ncols + col];
            acc[t] = __builtin_amdgcn_wmma_f32_16x16x4_f32(
                false, a, false, b, (short)0, acc[t], false, false);
        }
    }

    // C/D layout: VGPR r -> row rowBase + 8*(lane>>4) + r, col = t*16 + (lane&15)
    const int mOut = rowBase + ((lane >> 4) << 3);
#pragma unroll
    for (int t = 0; t < NT; ++t) {
        const int col = t * 16 + m;
#pragma unroll
        for (int r = 0; r < 8; ++r) {
            const int rr = mOut + r;
            if (rr < nRows) H[(size_t)rr * ncols + col] = acc[t][r];
        }
    }
}

// ---------------- aggregation init: self-loop + bias ----------------
// out[i,c] = h[i,c] * dinv[i]^2 + bias[c]

template <int C>
__global__ void agg_init_kernel(const float* __restrict__ H,
                                const float* __restrict__ dinv,
                                const float* __restrict__ bias,
                                float* __restrict__ out, int n) {
    int idx = blockIdx.x * blockDim.x + threadIdx.x;
    if (idx < n * C) {
        const int i = idx / C;
        const int c = idx % C;
        const float di = dinv[i];
        out[idx] = H[idx] * di * di + bias[c];
    }
}

// ---------------- edge scatter: out[dst] += h[src] * (dinv[src]*w*dinv[dst]) ----
// C/4 lanes per edge, float4 gather (coalesced), 4 f32 atomics per lane.

template <int C>
__global__ void edge_scatter_kernel(const long long* __restrict__ src,
                                    const long long* __restrict__ dst,
                                    const float* __restrict__ w,
                                    const float* __restrict__ dinv,
                                    const float* __restrict__ H,
                                    float* __restrict__ out, int E) {
    const int LPE = C / 4;
    const int t = blockIdx.x * blockDim.x + threadIdx.x;
    const int e = t / LPE;
    const int l = t % LPE;
    if (e >= E) return;
    const int s = (int)src[e];
    const int d = (int)dst[e];
    const float norm = dinv[s] * w[e] * dinv[d];
    const float4 v = ((const float4*)(H + (size_t)s * C))[l];
    float* op = out + (size_t)d * C + l * 4;
    atomicAdd(op + 0, v.x * norm);
    atomicAdd(op + 1, v.y * norm);
    atomicAdd(op + 2, v.z * norm);
    atomicAdd(op + 3, v.w * norm);
}

// ---------------- launch ----------------

extern "C" void kernel_launch(void* const* d_in, const int* in_sizes, int n_in,
                              void* d_out, int out_size, void* d_ws, size_t ws_size,
                              hipStream_t stream) {
    const float*     x   = (const float*)d_in[0];
    const long long* ei  = (const long long*)d_in[1];   // int64 edge_index [2, E]
    const float*     ew  = (const float*)d_in[2];
    const float*     W1  = (const float*)d_in[3];
    const float*     b1  = (const float*)d_in[4];
    const float*     W2  = (const float*)d_in[5];
    const float*     b2  = (const float*)d_in[6];
    float*           out = (float*)d_out;

    const int cHid = in_sizes[4];            // 64
    const int cOut = in_sizes[6];            // 32
    const int cIn  = in_sizes[3] / cHid;     // 64
    const int n    = in_sizes[0] / cIn;      // 100000
    const int E    = in_sizes[2];            // 1200000

    const long long* src = ei;
    const long long* dst = ei + E;

    // workspace carve-up (floats): dinv[n] | h[n*cHid] | agg1[n*cHid]
    float* dinv = (float*)d_ws;
    float* h    = dinv + n;                  // reused for h1 (n x 64) and h2 (n x 32)
    float* agg1 = h + (size_t)n * cHid;

    const int B = 256;

    // normalization (computed once, shared by both layers)
    deg_init_kernel<<<(n + B - 1) / B, B, 0, stream>>>(dinv, n);
    deg_acc_kernel<<<(E + B - 1) / B, B, 0, stream>>>(dst, ew, dinv, E);
    deg_rsqrt_kernel<<<(n + B - 1) / B, B, 0, stream>>>(dinv, n);

    const int tiles  = (n + 15) / 16;
    const int gBlk   = (tiles + 7) / 8;      // 8 waves (256 threads) per block

    // layer 1: h1 = x @ W1
    gemm_wmma_kernel<4><<<gBlk, B, 0, stream>>>(x, W1, h, n, cIn, 0);
    // agg1 = self-loop + bias, then scatter edges
    agg_init_kernel<64><<<((size_t)n * 64 + B - 1) / B, B, 0, stream>>>(h, dinv, b1, agg1, n);
    edge_scatter_kernel<64><<<((size_t)E * 16 + B - 1) / B, B, 0, stream>>>(
        src, dst, ew, dinv, h, agg1, E);

    // layer 2: h2 = relu(agg1) @ W2   (ReLU fused into A-operand load)
    gemm_wmma_kernel<2><<<gBlk, B, 0, stream>>>(agg1, W2, h, n, cHid, 1);
    agg_init_kernel<32><<<((size_t)n * 32 + B - 1) / B, B, 0, stream>>>(h, dinv, b2, out, n);
    edge_scatter_kernel<32><<<((size_t)E * 8 + B - 1) / B, B, 0, stream>>>(
        src, dst, ew, dinv, h, out, E);
}